// SimpleGCN_59227599011900
// MI455X (gfx1250) — compile-verified
//
#include <hip/hip_runtime.h>
#include <hip/hip_bf16.h>

typedef __attribute__((ext_vector_type(2))) float v2f;
typedef __attribute__((ext_vector_type(8))) float v8f;

#define F 128            // feature width (HL1 == F_IN == 128)
#define GFEAT 16
#define GHID 32

static inline int ceil_div(long long a, long long b) { return (int)((a + b - 1) / b); }

// ---------------------------------------------------------------------------
// zero fill
// ---------------------------------------------------------------------------
__global__ void k_zero(float* __restrict__ p, long long n) {
    long long i = (long long)blockIdx.x * blockDim.x + threadIdx.x;
    if (i < n) p[i] = 0.0f;
}

// ---------------------------------------------------------------------------
// degree accumulation (dst side), then dinv = 1/sqrt(deg + 1)
// ---------------------------------------------------------------------------
__global__ void k_degree(const long long* __restrict__ dst, float* __restrict__ deg, int E) {
    int e = blockIdx.x * blockDim.x + threadIdx.x;
    if (e < E) atomicAdd(&deg[dst[e]], 1.0f);
}

__global__ void k_dinv(float* __restrict__ deg, int n) {
    int i = blockIdx.x * blockDim.x + threadIdx.x;
    if (i < n) deg[i] = rsqrtf(deg[i] + 1.0f);   // deg+1 >= 1 always
}

// ---------------------------------------------------------------------------
// out[nrows,128] = in[nrows,128] @ W[128,128]  — fp32 WMMA 16x16x4
// block = 256 threads (8 waves). Block owns 16 rows; wave w owns cols 16w..16w+15.
// A tile (16x128) staged in LDS; B fragments streamed from global (L2-resident W).
// ---------------------------------------------------------------------------
__global__ __launch_bounds__(256) void k_gemm128(const float* __restrict__ in,
                                                 const float* __restrict__ W,
                                                 float* __restrict__ out,
                                                 int nrows) {
    __shared__ float As[16 * F];
    const int rowbase = blockIdx.x * 16;
    const int tid = threadIdx.x;

    // cooperative A-tile load: 2048 floats, float4 per thread x2
    for (int i = tid; i < (16 * F) / 4; i += 256) {
        int r = (i * 4) / F, c = (i * 4) % F;
        float4 v = make_float4(0.f, 0.f, 0.f, 0.f);
        int row = rowbase + r;
        if (row < nrows) v = *(const float4*)(in + (size_t)row * F + c);
        *(float4*)(As + r * F + c) = v;
    }
    __syncthreads();

    const int wave    = tid >> 5;        // 0..7 -> column tile
    const int lane    = tid & 31;
    const int colbase = wave * 16;
    const int lhalf   = lane >> 4;       // 0: K even pair base, 1: K odd pair base
    const int l15     = lane & 15;

    v8f acc = {};
    for (int k = 0; k < F; k += 4) {
        // A 16x4 f32 frag: VGPR0 = K {0|2}, VGPR1 = K {1|3} by lane half (ISA 7.12.2)
        v2f a, b;
        a.x = As[l15 * F + k + lhalf * 2 + 0];
        a.y = As[l15 * F + k + lhalf * 2 + 1];
        // B 4x16 f32 frag, mirrored layout: row K striped across lanes
        b.x = W[(size_t)(k + lhalf * 2 + 0) * F + colbase + l15];
        b.y = W[(size_t)(k + lhalf * 2 + 1) * F + colbase + l15];
        acc = __builtin_amdgcn_wmma_f32_16x16x4_f32(
            /*neg_a=*/false, a, /*neg_b=*/false, b,
            /*c_mod=*/(short)0, acc, /*reuse_a=*/false, /*reuse_b=*/false);
    }

    // C/D layout: VGPR j -> (M=j, N=l15) lanes 0-15 ; (M=j+8, N=l15) lanes 16-31
    for (int j = 0; j < 8; ++j) {
        int row = rowbase + j + lhalf * 8;
        if (row < nrows) out[(size_t)row * F + colbase + l15] = acc[j];
    }
}

// ---------------------------------------------------------------------------
// edge-parallel normalized scatter-add: one wave32 per edge, float4 per lane
// agg[dst] += h[src] * dinv[src] * dinv[dst]
// ---------------------------------------------------------------------------
__global__ __launch_bounds__(256) void k_aggregate(const float* __restrict__ h,
                                                   const long long* __restrict__ src,
                                                   const long long* __restrict__ dst,
                                                   const float* __restrict__ dinv,
                                                   float* __restrict__ agg, int E) {
    long long t = (long long)blockIdx.x * blockDim.x + threadIdx.x;
    int e = (int)(t >> 5);
    int lane = (int)(t & 31);
    if (e >= E) return;
    long long s = src[e], d = dst[e];
    float nrm = dinv[s] * dinv[d];
    float4 hv = *(const float4*)(h + (size_t)s * F + lane * 4);
    float* ap = agg + (size_t)d * F + lane * 4;
    atomicAdd(ap + 0, hv.x * nrm);
    atomicAdd(ap + 1, hv.y * nrm);
    atomicAdd(ap + 2, hv.z * nrm);
    atomicAdd(ap + 3, hv.w * nrm);
}

// ---------------------------------------------------------------------------
// finalize (in place on agg): agg = relu(agg + h * dinv^2 + bias)
// ---------------------------------------------------------------------------
__global__ __launch_bounds__(256) void k_finalize(float* __restrict__ agg,
                                                  const float* __restrict__ h,
                                                  const float* __restrict__ dinv,
                                                  const float* __restrict__ bias,
                                                  int n) {
    long long t = (long long)blockIdx.x * blockDim.x + threadIdx.x;
    int i = (int)(t >> 5);
    int lane = (int)(t & 31);
    if (i >= n) return;
    float di = dinv[i];
    float sl = di * di;
    float4 a  = *(float4*)(agg + (size_t)i * F + lane * 4);
    float4 hv = *(const float4*)(h + (size_t)i * F + lane * 4);
    float4 b  = *(const float4*)(bias + lane * 4);
    a.x = fmaxf(fmaf(hv.x, sl, a.x) + b.x, 0.0f);
    a.y = fmaxf(fmaf(hv.y, sl, a.y) + b.y, 0.0f);
    a.z = fmaxf(fmaf(hv.z, sl, a.z) + b.z, 0.0f);
    a.w = fmaxf(fmaf(hv.w, sl, a.w) + b.w, 0.0f);
    *(float4*)(agg + (size_t)i * F + lane * 4) = a;
}

// ---------------------------------------------------------------------------
// per-graph sum pooling via atomics
// ---------------------------------------------------------------------------
__global__ __launch_bounds__(256) void k_pool(const float* __restrict__ hfin,
                                              const int* __restrict__ batch,
                                              float* __restrict__ pooled, int n) {
    long long t = (long long)blockIdx.x * blockDim.x + threadIdx.x;
    int i = (int)(t >> 5);
    int lane = (int)(t & 31);
    if (i >= n) return;
    int g = batch[i];
    float4 hv = *(const float4*)(hfin + (size_t)i * F + lane * 4);
    float* pp = pooled + (size_t)g * F + lane * 4;
    atomicAdd(pp + 0, hv.x);
    atomicAdd(pp + 1, hv.y);
    atomicAdd(pp + 2, hv.z);
    atomicAdd(pp + 3, hv.w);
}

// ---------------------------------------------------------------------------
// MLP head: one block (128 threads) per graph
// ---------------------------------------------------------------------------
__global__ __launch_bounds__(128) void k_head(const float* __restrict__ pooled,
                                              const float* __restrict__ gfs,
                                              const float* __restrict__ g1W,
                                              const float* __restrict__ g1b,
                                              const float* __restrict__ l1W,
                                              const float* __restrict__ l1b,
                                              const float* __restrict__ l2W,
                                              const float* __restrict__ l2b,
                                              const float* __restrict__ l3W,
                                              const float* __restrict__ l3b,
                                              const float* __restrict__ l4W,
                                              const float* __restrict__ l4b,
                                              float* __restrict__ out) {
    __shared__ float z[F + GHID];   // concat buffer
    __shared__ float y[F];
    __shared__ float red[4];
    const int g = blockIdx.x;
    const int c = threadIdx.x;

    // y1 = relu(pooled[g] @ lin1_W + lin1_b) -> z[0:128]
    {
        float acc = l1b[c];
        const float* p = pooled + (size_t)g * F;
        for (int k = 0; k < F; ++k) acc = fmaf(p[k], l1W[k * F + c], acc);
        z[c] = fmaxf(acc, 0.0f);
    }
    // g1 = relu(gfs[g] @ global1_W + global1_b) -> z[128:160]
    if (c < GHID) {
        float acc = g1b[c];
        const float* gf = gfs + g * GFEAT;
        for (int k = 0; k < GFEAT; ++k) acc = fmaf(gf[k], g1W[k * GHID + c], acc);
        z[F + c] = fmaxf(acc, 0.0f);
    }
    __syncthreads();
    // y2 = relu(z @ lin2_W[160,128] + lin2_b)
    {
        float acc = l2b[c];
        for (int k = 0; k < F + GHID; ++k) acc = fmaf(z[k], l2W[k * F + c], acc);
        y[c] = fmaxf(acc, 0.0f);
    }
    __syncthreads();
    // y3 = relu(y @ lin3_W + lin3_b) -> z[0:128]
    {
        float acc = l3b[c];
        for (int k = 0; k < F; ++k) acc = fmaf(y[k], l3W[k * F + c], acc);
        z[c] = fmaxf(acc, 0.0f);
    }
    __syncthreads();
    // out[g] = z @ lin4_W[128,1] + lin4_b
    float v = z[c] * l4W[c];
    for (int off = 16; off > 0; off >>= 1) v += __shfl_down(v, off, 32);
    if ((c & 31) == 0) red[c >> 5] = v;
    __syncthreads();
    if (c == 0) out[g] = red[0] + red[1] + red[2] + red[3] + l4b[0];
}

// ---------------------------------------------------------------------------
extern "C" void kernel_launch(void* const* d_in, const int* in_sizes, int n_in,
                              void* d_out, int out_size, void* d_ws, size_t ws_size,
                              hipStream_t stream) {
    const float*     x     = (const float*)d_in[0];
    const long long* ei    = (const long long*)d_in[1];
    const int*       batch = (const int*)d_in[2];
    const float*     gfs   = (const float*)d_in[3];
    // d_in[4] = num_graphs scalar (on device) — derive G from sizes instead
    const float* convW[4] = { (const float*)d_in[5], (const float*)d_in[7],
                              (const float*)d_in[9], (const float*)d_in[11] };
    const float* convB[4] = { (const float*)d_in[6], (const float*)d_in[8],
                              (const float*)d_in[10], (const float*)d_in[12] };
    const float* g1W = (const float*)d_in[13];
    const float* g1b = (const float*)d_in[14];
    const float* l1W = (const float*)d_in[15];
    const float* l1b = (const float*)d_in[16];
    const float* l2W = (const float*)d_in[17];
    const float* l2b = (const float*)d_in[18];
    const float* l3W = (const float*)d_in[19];
    const float* l3b = (const float*)d_in[20];
    const float* l4W = (const float*)d_in[21];
    const float* l4b = (const float*)d_in[22];
    float* out = (float*)d_out;

    const int N = in_sizes[0] / F;
    const int E = in_sizes[1] / 2;
    const int G = in_sizes[3] / GFEAT;
    const long long NF = (long long)N * F;

    const long long* src = ei;
    const long long* dst = ei + E;

    // ws layout (floats), 256B aligned slices
    float* ws = (float*)d_ws;
    size_t off = 0;
    float* dinv = ws + off;  off += ((size_t)N + 63) & ~(size_t)63;
    float* hbuf = ws + off;  off += (size_t)NF;        // GEMM output per layer
    float* abuf = ws + off;  off += (size_t)NF;        // agg / layer output (in place)
    float* pooled = ws + off; off += (size_t)G * F;
    (void)ws_size; (void)n_in; (void)out_size;

    // 1) degree -> dinv
    k_zero<<<ceil_div(N, 256), 256, 0, stream>>>(dinv, N);
    k_degree<<<ceil_div(E, 256), 256, 0, stream>>>(dst, dinv, E);
    k_dinv<<<ceil_div(N, 256), 256, 0, stream>>>(dinv, N);

    // 2) 4 GCN layers; layer input = x for l0, else abuf (finalized in place)
    const int gemmBlocks = ceil_div(N, 16);
    const int edgeBlocks = ceil_div((long long)E * 32, 256);
    const int nodeBlocks = ceil_div((long long)N * 32, 256);
    for (int l = 0; l < 4; ++l) {
        const float* in = (l == 0) ? x : abuf;
        k_gemm128<<<gemmBlocks, 256, 0, stream>>>(in, convW[l], hbuf, N);
        k_zero<<<ceil_div(NF, 256), 256, 0, stream>>>(abuf, NF);
        k_aggregate<<<edgeBlocks, 256, 0, stream>>>(hbuf, src, dst, dinv, abuf, E);
        k_finalize<<<nodeBlocks, 256, 0, stream>>>(abuf, hbuf, dinv, convB[l], N);
    }

    // 3) pooling
    k_zero<<<ceil_div((long long)G * F, 256), 256, 0, stream>>>(pooled, (long long)G * F);
    k_pool<<<nodeBlocks, 256, 0, stream>>>(abuf, batch, pooled, N);

    // 4) MLP head
    k_head<<<G, 128, 0, stream>>>(pooled, gfs, g1W, g1b, l1W, l1b,
                                  l2W, l2b, l3W, l3b, l4W, l4b, out);
}